// COPNLL_10737418240820
// MI455X (gfx1250) — compile-verified
//
#include <hip/hip_runtime.h>
#include <hip/hip_bf16.h>

// ---------------------------------------------------------------------------
// GLMM Bernoulli NLL with 5-pt Gauss-Hermite quadrature, MI455X (gfx1250).
// Phase 1: zero 7*G f32 counters in d_ws (+ d_out).
// Phase 2: streaming scatter: 7 global_atomic_add_f32 per element (L2-resident
//          280KB counter set). Softplus on hardware transcendentals
//          (v_exp_f32 / v_log_f32) so the loop is atomic-bound, not VALU-bound.
// Phase 3: per-group LSE + global sum; intra-wave reduction done with
//          V_WMMA_F32_16X16X4_F32 (ones-B trick, exact f32).
// ---------------------------------------------------------------------------

#define NGQ 5

__device__ __constant__ float c_xk[NGQ] = {
    -2.0201828704560856f, -0.9585724646138185f, 0.0f,
     0.9585724646138185f,  2.0201828704560856f };
__device__ __constant__ float c_wk[NGQ] = {
     0.019953242059045913f, 0.3936193231522412f, 0.9453087204829419f,
     0.3936193231522412f,   0.019953242059045913f };

typedef __attribute__((ext_vector_type(2))) float v2f;
typedef __attribute__((ext_vector_type(8))) float v8f;

__device__ __forceinline__ void atomAddF32(float* p, float v) {
    // relaxed agent-scope fadd -> global_atomic_add_f32 (no CAS loop)
    __hip_atomic_fetch_add(p, v, __ATOMIC_RELAXED, __HIP_MEMORY_SCOPE_AGENT);
}

// Exact f32 sum of the 32 lane values of a wave via V_WMMA_F32_16X16X4_F32.
// A (16x4): lanes 0-15 supply K=0 (VGPR0), lanes 16-31 supply K=2; the other
// K slots are zeroed. B = ones. D[m][*] = v[m] + v[m+16]; lane n<16 holds
// D[j][n] (j = VGPR) and lane 16+n holds D[j+8][n], so summing the 8 D VGPRs
// gives each half-wave the sum of 16 values; one xor-16 shuffle finishes.
// Requires EXEC == all ones at the call site.
__device__ __forceinline__ float wave_sum_wmma(float v) {
    v2f a; a[0] = v;    a[1] = 0.0f;
    v2f b; b[0] = 1.0f; b[1] = 1.0f;
    v8f c = {};
    v8f d = __builtin_amdgcn_wmma_f32_16x16x4_f32(
        /*neg_a=*/false, a, /*neg_b=*/false, b,
        /*c_mod=*/(short)0, c, /*reuse_a=*/false, /*reuse_b=*/false);
    float h = ((d[0] + d[1]) + (d[2] + d[3])) + ((d[4] + d[5]) + (d[6] + d[7]));
    h += __shfl_xor(h, 16, 32);
    return h;   // full-wave sum, broadcast to all lanes
}

// Stable softplus log1p(exp(z)) on hardware transcendentals:
//   softplus(z) = max(z,0) + ln2 * v_log_f32(1 + v_exp_f32(-|z|*log2e))
// One v_exp_f32 + one v_log_f32 + FMAs; ~1e-7 rel error, far below the
// f32 atomic-ordering noise of the segment sums.
__device__ __forceinline__ float softplus_fast(float z) {
    const float LOG2E = 1.4426950408889634f;
    const float LN2   = 0.6931471805599453f;
    const float t = __builtin_amdgcn_exp2f(-fabsf(z) * LOG2E);   // exp(-|z|)
    return fmaxf(z, 0.0f) + __builtin_amdgcn_logf(1.0f + t) * LN2;
}

// -------------------------------- kernels ----------------------------------

__global__ void glmm_zero_kernel(const int* __restrict__ ng,
                                 float* __restrict__ ws,
                                 float* __restrict__ out) {
    const int G = *ng;
    const int total = 7 * G;
    const int stride = gridDim.x * blockDim.x;
    for (int i = blockIdx.x * blockDim.x + threadIdx.x; i < total; i += stride)
        ws[i] = 0.0f;
    if (blockIdx.x == 0 && threadIdx.x == 0) out[0] = 0.0f;
}

__global__ void glmm_scatter_kernel(const float* __restrict__ y,
                                    const float* __restrict__ f,
                                    const float* __restrict__ sig2b,
                                    const int*   __restrict__ zidx,
                                    const int*   __restrict__ ng,
                                    float* __restrict__ ws,
                                    int n) {
    const int i = blockIdx.x * blockDim.x + threadIdx.x;
    if (i >= n) return;

    // stream-ahead prefetch (global_prefetch_b8); clamp to stay in range
    const int pf = (i + 4096 < n) ? (i + 4096) : (n - 1);
    __builtin_prefetch(&y[pf], 0, 1);
    __builtin_prefetch(&f[pf], 0, 1);
    __builtin_prefetch(&zidx[pf], 0, 1);

    const int   G      = *ng;
    const float cscale = sqrtf(2.0f * (*sig2b));

    const float yi = y[i];
    const float fi = f[i];
    const int   g  = zidx[i];

    float* __restrict__ yf_g = ws;
    float* __restrict__ ys_g = ws + G;
    float* __restrict__ lg   = ws + 2 * G;   // lg[k*G + g]

    atomAddF32(&yf_g[g], yi * fi);
    atomAddF32(&ys_g[g], yi);

#pragma unroll
    for (int k = 0; k < NGQ; ++k) {
        const float z = fi + cscale * c_xk[k];
        atomAddF32(&lg[k * G + g], softplus_fast(z));
    }
}

__global__ void glmm_finalize_kernel(const float* __restrict__ sig2b,
                                     const int*   __restrict__ ng,
                                     const float* __restrict__ ws,
                                     float* __restrict__ out) {
    const int   G      = *ng;
    const float cscale = sqrtf(2.0f * (*sig2b));
    const float inv_sqrt_pi = 0.5641895835477563f;

    const float* __restrict__ yf_g = ws;
    const float* __restrict__ ys_g = ws + G;
    const float* __restrict__ lg   = ws + 2 * G;

    const int stride = gridDim.x * blockDim.x;
    const int tid    = blockIdx.x * blockDim.x + threadIdx.x;
    // wave-uniform trip count (Gpad multiple of stride) so lanes reconverge
    const int Gpad   = ((G + stride - 1) / stride) * stride;

    float acc = 0.0f;
    for (int i = tid; i < Gpad; i += stride) {
        const int   gi     = (i < G) ? i : (G - 1);   // clamped, non-divergent
        const float active = (i < G) ? 1.0f : 0.0f;

        const float yf = yf_g[gi];
        const float ys = ys_g[gi];

        float e[NGQ];
        float m = -3.402823e38f;
#pragma unroll
        for (int k = 0; k < NGQ; ++k) {
            const float ck = cscale * c_xk[k];
            e[k] = yf + ys * ck - lg[k * G + gi];
            m = fmaxf(m, e[k]);
        }
        float ksum = 0.0f;
#pragma unroll
        for (int k = 0; k < NGQ; ++k)
            ksum += expf(e[k] - m) * (c_wk[k] * inv_sqrt_pi);   // precise: 10k groups only

        acc += active * (-(m + logf(ksum)));
    }

    // intra-wave exact f32 reduction on the matrix pipe (EXEC full here)
    const float wsum = wave_sum_wmma(acc);

    __shared__ float sh[8];
    const int lane = threadIdx.x & 31;
    const int wid  = threadIdx.x >> 5;
    if (lane == 0) sh[wid] = wsum;
    __syncthreads();
    if (threadIdx.x == 0) {
        float b = 0.0f;
        const int nw = blockDim.x >> 5;
        for (int w = 0; w < nw; ++w) b += sh[w];
        atomAddF32(out, b);
    }
}

// ------------------------------- launcher ----------------------------------

extern "C" void kernel_launch(void* const* d_in, const int* in_sizes, int n_in,
                              void* d_out, int out_size, void* d_ws, size_t ws_size,
                              hipStream_t stream) {
    const float* y     = (const float*)d_in[0];   // y_true  (N)
    const float* f     = (const float*)d_in[1];   // y_pred  (N)
    const float* sig2b = (const float*)d_in[2];   // scalar
    const int*   zidx  = (const int*)  d_in[3];   // Z_idx   (N)
    const int*   ng    = (const int*)  d_in[4];   // n_groups (scalar)
    float*       out   = (float*)d_out;
    float*       ws    = (float*)d_ws;            // 7*G floats

    const int n = in_sizes[0];

    glmm_zero_kernel<<<128, 256, 0, stream>>>(ng, ws, out);
    glmm_scatter_kernel<<<(n + 255) / 256, 256, 0, stream>>>(y, f, sig2b, zidx, ng, ws, n);
    glmm_finalize_kernel<<<80, 256, 0, stream>>>(sig2b, ng, ws, out);
}